// TransOp_expm_47261820125388
// MI455X (gfx1250) — compile-verified
//
#include <hip/hip_runtime.h>

// CDNA5 / gfx1250: wave32, WMMA f32 16x16x4.
typedef float v2f __attribute__((ext_vector_type(2)));
typedef float v8f __attribute__((ext_vector_type(8)));

// 3x3 row-major matmul: d = a * b
__device__ __forceinline__ void mm3(const float* __restrict__ a,
                                    const float* __restrict__ b,
                                    float* __restrict__ d) {
#pragma unroll
  for (int i = 0; i < 3; ++i) {
#pragma unroll
    for (int j = 0; j < 3; ++j) {
      float acc = a[i * 3 + 0] * b[0 * 3 + j];
      acc = fmaf(a[i * 3 + 1], b[1 * 3 + j], acc);
      acc = fmaf(a[i * 3 + 2], b[2 * 3 + j], acc);
      d[i * 3 + j] = acc;
    }
  }
}

// One batch element per lane. Each wave:
//   - two WMMA_F32_16X16X4 tile pairs compute T = c @ psi for 32 elements
//   - bounce 16x16 accumulators through LDS so each lane owns its 3x3 T
//   - per-lane scaling-and-squaring expm (Taylor-8 Horner) + expT @ x
__global__ __launch_bounds__(256) void transop_expm_kernel(
    const float* __restrict__ c,    // (B, 6)
    const float* __restrict__ x,    // (B, 3)
    const float* __restrict__ psi,  // (6, 9)
    float* __restrict__ out,        // (B, 3)
    int nB) {
  __shared__ float psiLds[8][16];        // K rows 0..7 (6,7 zero), N padded to 16
  __shared__ float tLds[8][2][16][16];   // [wave][tile][M][N] = 16 KB

  const int tid  = threadIdx.x;
  const int wave = tid >> 5;
  const int lane = tid & 31;
  const int l15  = lane & 15;   // A: M row / B,C: N column
  const int hi   = lane >> 4;   // upper half-wave selector

  // Zero-padded psi staged once per block (branchless B-fragment gathers).
  if (tid < 128) {
    const int k = tid >> 4, n = tid & 15;
    psiLds[k][n] = (k < 6 && n < 9) ? psi[k * 9 + n] : 0.0f;
  }
  __syncthreads();

  const long bbase = (long)blockIdx.x * 256 + (long)wave * 32;

  // B fragments (4x16 f32 per WMMA): VGPR0 = rows {0,2}(K-block0)/{4,6}(K-block1),
  // VGPR1 = rows {1,3}/{5,7}; lanes 0-15 low K pair, 16-31 high K pair.
  v2f bk0, bk1;
  bk0.x = psiLds[hi ? 2 : 0][l15];
  bk0.y = psiLds[hi ? 3 : 1][l15];
  bk1.x = psiLds[hi ? 6 : 4][l15];
  bk1.y = psiLds[hi ? 7 : 5][l15];

  // Data-side zeroing mask for the K=6/7 pad rows of A (keeps loads
  // unconditional: no divergent s_and_saveexec around global_load_b32).
  const float loMask = hi ? 0.0f : 1.0f;

#pragma unroll
  for (int t = 0; t < 2; ++t) {
    long br = bbase + t * 16 + l15;            // batch row for this A-matrix row
    if (br > (long)nB - 1) br = (long)nB - 1;  // clamp (keeps EXEC all-ones)
    const float* crow = c + br * 6;

    // A fragments (16x4 f32): VGPR0 = K{0|2} / K{4|6}, VGPR1 = K{1|3} / K{5|7}.
    v2f ak0, ak1;
    ak0.x = crow[hi ? 2 : 0];
    ak0.y = crow[hi ? 3 : 1];
    ak1.x = crow[4] * loMask;   // K=6 pad on upper half-wave
    ak1.y = crow[5] * loMask;   // K=7 pad on upper half-wave

    v8f acc = {};
    acc = __builtin_amdgcn_wmma_f32_16x16x4_f32(false, ak0, false, bk0,
                                                (short)0, acc, false, false);
    acc = __builtin_amdgcn_wmma_f32_16x16x4_f32(false, ak1, false, bk1,
                                                (short)0, acc, false, false);

    // C/D layout: VGPR v holds M = v + 8*hi, N = l15.
#pragma unroll
    for (int v = 0; v < 8; ++v)
      tLds[wave][t][v + hi * 8][l15] = acc[v];
  }
  __syncthreads();

  const long b = bbase + lane;  // this lane's batch element: tile=hi, row=l15

  float A[9];
#pragma unroll
  for (int i = 0; i < 9; ++i) A[i] = tLds[wave][hi][l15][i];

  // --- expm via scaling and squaring ---
  float fro2 = 0.0f;
#pragma unroll
  for (int i = 0; i < 9; ++i) fro2 = fmaf(A[i], A[i], fro2);
  const float fro = sqrtf(fro2);

  int s = 0;
  if (fro > 0.25f) {
    int e;
    (void)frexpf(fro, &e);       // fro = m * 2^e, m in [0.5, 1)
    s = e + 2;                   // scaled norm = m * 2^-2 < 0.25
    if (s > 30) s = 30;
    const float sc = ldexpf(1.0f, -s);
#pragma unroll
    for (int i = 0; i < 9; ++i) A[i] *= sc;
  }

  // Taylor-8 Horner: R = I + A(I + A/2(I + ... (I + A/8)))
  float R[9] = {1.f, 0.f, 0.f, 0.f, 1.f, 0.f, 0.f, 0.f, 1.f};
#pragma unroll
  for (int k = 8; k >= 1; --k) {
    float P[9];
    mm3(A, R, P);
    const float inv = 1.0f / (float)k;
#pragma unroll
    for (int i = 0; i < 9; ++i) R[i] = P[i] * inv;
    R[0] += 1.0f; R[4] += 1.0f; R[8] += 1.0f;
  }

  // Undo the scaling: square s times.
  for (int i = 0; i < s; ++i) {
    float P[9];
    mm3(R, R, P);
#pragma unroll
    for (int j = 0; j < 9; ++j) R[j] = P[j];
  }

  if (b < (long)nB) {
    const float x0 = x[b * 3 + 0];
    const float x1 = x[b * 3 + 1];
    const float x2 = x[b * 3 + 2];
#pragma unroll
    for (int i = 0; i < 3; ++i)
      out[b * 3 + i] =
          fmaf(R[i * 3 + 0], x0, fmaf(R[i * 3 + 1], x1, R[i * 3 + 2] * x2));
  }
}

extern "C" void kernel_launch(void* const* d_in, const int* in_sizes, int n_in,
                              void* d_out, int out_size, void* d_ws,
                              size_t ws_size, hipStream_t stream) {
  const float* c   = (const float*)d_in[0];  // (B, 6)
  const float* x   = (const float*)d_in[1];  // (B, 3, 1)
  const float* psi = (const float*)d_in[2];  // (6, 3, 3)
  float* out = (float*)d_out;                // (B, 3, 1)

  const int nB = in_sizes[0] / 6;
  const int blocks = (nB + 255) / 256;       // 256 batch elements per block
  transop_expm_kernel<<<dim3(blocks), dim3(256), 0, stream>>>(c, x, psi, out,
                                                              nB);
}